// SpecialSpmm_81277961109513
// MI455X (gfx1250) — compile-verified
//
#include <hip/hip_runtime.h>
#include <stdint.h>

// COO SpMM: out[row[e]] += values[e] * b[col[e]]  (D = 128 f32 features)
//
// MI455X strategy: bandwidth/atomic bound (AI ~ 0.25 flop/B). b (51.2MB) and
// out (51.2MB) both fit in the 192MB L2, so gathers and scatter-atomics run at
// L2 speed. One wave32 per edge: 32 lanes x 16B = one 512B feature row.
// Async global->LDS copy engine double-buffers the gather (ASYNCcnt),
// native f32 L2 atomics do the scatter (no-return form, STOREcnt).

#define BLOCK 256
#define WAVES_PER_BLOCK (BLOCK / 32)
#define D_FEAT 128
#define ROW_BYTES 512  // 128 * 4B

__global__ __launch_bounds__(BLOCK) void SpecialSpmm_coo_async_kernel(
    const int* __restrict__ rowIdx,
    const int* __restrict__ colIdx,
    const float* __restrict__ vals,
    const float* __restrict__ bmat,
    float* __restrict__ out,
    int nEdges)
{
    // wave-private double buffers: 8 waves * 2 * 512B = 8KB LDS per block
    __shared__ __align__(16) float smem[WAVES_PER_BLOCK * 2 * D_FEAT];

    const int lane = threadIdx.x & 31;
    // force wave id into an SGPR so per-edge index/value loads become s_load
    const int wid = __builtin_amdgcn_readfirstlane((int)(threadIdx.x >> 5));
    const int gwave  = (int)blockIdx.x * WAVES_PER_BLOCK + wid;
    const int nwaves = (int)gridDim.x * WAVES_PER_BLOCK;

    int e = gwave;
    if (e >= nEdges) return;

    // LDS byte addresses for this wave's two buffers (lane-sliced, 16B/lane).
    // Low 32 bits of the generic pointer are the LDS byte offset.
    const uint32_t laneB   = (uint32_t)lane * 16u;
    const uint32_t ldsWave = (uint32_t)(uintptr_t)(&smem[wid * 2 * D_FEAT]);
    const uint32_t ldsBuf[2] = { ldsWave + laneB, ldsWave + ROW_BYTES + laneB };

    // ---- prologue: stage first edge's b-row into buffer 0 (async copy) ----
    {
        const uint32_t c    = (uint32_t)colIdx[e];
        const uint32_t voff = c * (uint32_t)ROW_BYTES + laneB;
        asm volatile("global_load_async_to_lds_b128 %0, %1, %2"
                     :: "v"(ldsBuf[0]), "v"(voff), "s"(bmat)
                     : "memory");
    }

    int cur = 0;
    for (; e < nEdges; e += nwaves) {
        const int  en      = e + nwaves;
        const bool hasNext = (en < nEdges);

        if (hasNext) {
            // Reused buffer was consumed (ds reads) last iteration; make sure
            // those LDS reads retired before the async engine overwrites it.
            const uint32_t c    = (uint32_t)colIdx[en];
            const uint32_t voff = c * (uint32_t)ROW_BYTES + laneB;
            asm volatile("s_wait_dscnt 0x0\n\t"
                         "global_load_async_to_lds_b128 %0, %1, %2"
                         :: "v"(ldsBuf[cur ^ 1]), "v"(voff), "s"(bmat)
                         : "memory");
            // two async ops in flight; <=1 guarantees the older one (cur) done
            asm volatile("s_wait_asynccnt 0x1" ::: "memory");
        } else {
            asm volatile("s_wait_asynccnt 0x0" ::: "memory");
        }

        const float    v = vals[e];                 // wave-uniform -> s_load
        const uint32_t r = (uint32_t)rowIdx[e];     // wave-uniform -> s_load

        // consume current buffer: ds_load_b128 per lane
        const float4 x = *reinterpret_cast<const float4*>(
            &smem[wid * 2 * D_FEAT + cur * D_FEAT + lane * 4]);

        float m0 = x.x * v;
        float m1 = x.y * v;
        float m2 = x.z * v;
        float m3 = x.w * v;

        // scatter: 4 native f32 atomics per lane, no-return form (STOREcnt)
        const uint32_t ooff = r * (uint32_t)ROW_BYTES + laneB;
        asm volatile("global_atomic_add_f32 %0, %1, %5\n\t"
                     "global_atomic_add_f32 %0, %2, %5 offset:4\n\t"
                     "global_atomic_add_f32 %0, %3, %5 offset:8\n\t"
                     "global_atomic_add_f32 %0, %4, %5 offset:12"
                     :: "v"(ooff), "v"(m0), "v"(m1), "v"(m2), "v"(m3), "s"(out)
                     : "memory");

        cur ^= 1;
    }
    // s_endpgm implicitly waits for outstanding stores/atomics
}

extern "C" void kernel_launch(void* const* d_in, const int* in_sizes, int n_in,
                              void* d_out, int out_size, void* d_ws, size_t ws_size,
                              hipStream_t stream) {
    (void)n_in; (void)d_ws; (void)ws_size;

    // setup_inputs order: indices (2,E) i32, values (E,) f32, n (scalar), b (N,128) f32
    const int*   indices = (const int*)d_in[0];
    const float* vals    = (const float*)d_in[1];
    const float* bmat    = (const float*)d_in[3];
    float*       out     = (float*)d_out;

    const int E = in_sizes[1];          // values count == #edges
    const int* rowIdx = indices;        // indices[0, :]
    const int* colIdx = indices + E;    // indices[1, :]

    // segment_sum starts from zero; harness poisons d_out, so clear every call
    hipMemsetAsync(d_out, 0, (size_t)out_size * sizeof(float), stream);

    const int blocks = 4096;            // 32768 waves ~ 98 edges/wave
    SpecialSpmm_coo_async_kernel<<<blocks, BLOCK, 0, stream>>>(
        rowIdx, colIdx, vals, bmat, out, E);
}